// AnchorBgBox_34007551050566
// MI455X (gfx1250) — compile-verified
//
#include <hip/hip_runtime.h>

typedef float v2f __attribute__((ext_vector_type(2)));
typedef float v8f __attribute__((ext_vector_type(8)));

#define N_TARGETS          64
#define ANCHORS_PER_BLOCK  128
#define THREADS            256   // 8 waves (wave32)

// Each wave computes a 16-anchor x 64-target IoU tile.
// union's broadcast add (area_a[m] + area_t[n]) is computed with
// V_WMMA_F32_16X16X4_F32 (exact: products are x1.0), rest is VALU.
__global__ __launch_bounds__(THREADS)
void iou_wmma_kernel(const float* __restrict__ anchor,
                     const float* __restrict__ target,
                     float* __restrict__ out)
{
    __shared__ float4 sA[ANCHORS_PER_BLOCK];  // 2 KB
    __shared__ float4 sT[N_TARGETS];          // 1 KB

    const int t = threadIdx.x;
    const int blockAnchorBase = blockIdx.x * ANCHORS_PER_BLOCK;

    if (t < ANCHORS_PER_BLOCK) {
        sA[t] = ((const float4*)anchor)[blockAnchorBase + t];
    } else if (t < ANCHORS_PER_BLOCK + N_TARGETS) {
        sT[t - ANCHORS_PER_BLOCK] = ((const float4*)target)[t - ANCHORS_PER_BLOCK];
    }
    __syncthreads();

    const int lane = t & 31;
    const int wid  = t >> 5;        // wave id in block: 0..7
    const int hi   = lane >> 4;     // half-wave select (rows 0-7 vs 8-15)
    const int col  = lane & 15;     // tile column
    const int waveBase = wid * 16;  // anchor offset within block

    // Register the 8 anchor rows this half-wave owns (matches C/D VGPR layout).
    float4 A[8];
#pragma unroll
    for (int r = 0; r < 8; ++r)
        A[r] = sA[waveBase + hi * 8 + r];

    // A-matrix fragment: A[m][0]=area_a[m], A[m][1]=1, A[m][2..3]=0.
    // Lanes 0-15 carry K=0,1 (V0,V1); lanes 16-31 carry K=2,3 -> zeros.
    float4 own = sA[waveBase + col];
    float areaA = (own.z - own.x) * (own.w - own.y);
    v2f afrag;
    afrag.x = hi ? 0.0f : areaA;
    afrag.y = hi ? 0.0f : 1.0f;

    const int rowBase = (blockAnchorBase + waveBase + hi * 8) * N_TARGETS;

#pragma unroll
    for (int tile = 0; tile < 4; ++tile) {
        float4 T = sT[tile * 16 + col];
        float areaT = (T.z - T.x) * (T.w - T.y);

        // B-matrix fragment: B[0][n]=1, B[1][n]=area_t[n], B[2..3][n]=0.
        v2f bfrag;
        bfrag.x = hi ? 0.0f : 1.0f;
        bfrag.y = hi ? 0.0f : areaT;

        v8f c = {};
        // d[r] = area_a[m] + area_t[n], m = waveBase + 8*hi + r, n = 16*tile + col
        v8f d = __builtin_amdgcn_wmma_f32_16x16x4_f32(
            /*neg_a=*/false, afrag, /*neg_b=*/false, bfrag,
            /*c_mod=*/(short)0, c, /*reuse_a=*/false, /*reuse_b=*/false);

#pragma unroll
        for (int r = 0; r < 8; ++r) {
            float ix1 = fmaxf(A[r].x, T.x);
            float iy1 = fmaxf(A[r].y, T.y);
            float ix2 = fminf(A[r].z, T.z);
            float iy2 = fminf(A[r].w, T.w);
            float iw  = fmaxf(ix2 - ix1, 0.0f);
            float ih  = fmaxf(iy2 - iy1, 0.0f);
            float inter = iw * ih;
            float uni = fmaxf(d[r] - inter, 1e-10f);
            // fast reciprocal + one Newton-Raphson step (~0.5 ulp)
            float rc = __builtin_amdgcn_rcpf(uni);
            rc = rc * __builtin_fmaf(-uni, rc, 2.0f);
            out[rowBase + r * N_TARGETS + tile * 16 + col] = inter * rc;
        }
    }
}

extern "C" void kernel_launch(void* const* d_in, const int* in_sizes, int n_in,
                              void* d_out, int out_size, void* d_ws, size_t ws_size,
                              hipStream_t stream) {
    const float* anchor = (const float*)d_in[0];
    const float* target = (const float*)d_in[1];
    float* out = (float*)d_out;

    const int nAnchors = in_sizes[0] / 4;              // [N,4] corner boxes
    const int blocks = nAnchors / ANCHORS_PER_BLOCK;   // 1048576/128 = 8192, exact

    iou_wmma_kernel<<<blocks, THREADS, 0, stream>>>(anchor, target, out);
}